// PointCloudAligner_72911364817425
// MI455X (gfx1250) — compile-verified
//
#include <hip/hip_runtime.h>
#include <math.h>

// ---------------------------------------------------------------------------
// PointCloudAligner loss on MI455X (gfx1250, wave32)
//
// K=1 NN row-min decomposed so the per-row constant leaves the WMMA:
//   min_j d2[i][j] = ||tp_i||^2 + min_j( ||y_j||^2 - 2*tp_i.y_j )
//
// Each 16x16 tile of ( ||y_j||^2 - 2*tp_i.y_j ) is one V_WMMA_F32_16X16X4_F32
// with an inline-zero C accumulator:
//   A[i]   = [tp_x, tp_y, tp_z, 1.0]          (16x4 f32)
//   B[:,j] = [-2y_x, -2y_y, -2y_z, ||y_j||^2] (4x16 f32)
//   C      = 0  (inline constant SRC2)
// ---------------------------------------------------------------------------

typedef __attribute__((ext_vector_type(2))) float v2f;
typedef __attribute__((ext_vector_type(8))) float v8f;

#define NPTS 16384
#define MPTS 16384

// ---------------------------------------------------------------------------
// Kernel 1: pack points for WMMA tiles.
//   Apack[i] = (s*x + t, ||s*x + t||^2)   (.w used only in finalize)
//   Bpack[j] = (-2*y, ||y||^2)
// ---------------------------------------------------------------------------
__global__ void __launch_bounds__(256)
pca_prep_kernel(const float* __restrict__ src,
                const float* __restrict__ tgt,
                const float* __restrict__ scale,
                const float* __restrict__ trans,
                float4* __restrict__ Apack,
                float4* __restrict__ Bpack) {
    int i = blockIdx.x * blockDim.x + threadIdx.x;
    if (i >= NPTS) return;

    float s  = __expf(scale[0]);
    float tx = trans[0], ty = trans[1], tz = trans[2];

    float ax = fmaf(s, src[3 * i + 0], tx);
    float ay = fmaf(s, src[3 * i + 1], ty);
    float az = fmaf(s, src[3 * i + 2], tz);
    float asq = fmaf(ax, ax, fmaf(ay, ay, az * az));
    Apack[i] = make_float4(ax, ay, az, asq);

    float bx = tgt[3 * i + 0];
    float by = tgt[3 * i + 1];
    float bz = tgt[3 * i + 2];
    float bsq = fmaf(bx, bx, fmaf(by, by, bz * bz));
    Bpack[i] = make_float4(-2.0f * bx, -2.0f * by, -2.0f * bz, bsq);
}

// ---------------------------------------------------------------------------
// Kernel 2: brute-force K=1 NN via WMMA.
// One wave owns a 16-row A tile and sweeps all M targets in 16-column tiles.
// Grid exactly covers N, so EXEC is all-1s as WMMA requires.
//
// VGPR layouts (CDNA5 ISA):
//   A 16x4 f32 : lanes 0-15 -> rows 0..15, vgpr0=K0, vgpr1=K1
//                lanes 16-31 -> rows 0..15, vgpr0=K2, vgpr1=K3
//   B 4x16 f32 : mirror (lanes 0-15: K0/K1 of col=lane;
//                lanes 16-31: K2/K3 of col=lane-16)
//   D 16x16    : vgpr r = row r (lanes 0-15) / row r+8 (lanes 16-31)
// ---------------------------------------------------------------------------
__global__ void __launch_bounds__(256)
pca_nn_kernel(const float4* __restrict__ Apack,
              const float4* __restrict__ Bpack,
              float* __restrict__ rowmin) {
    const int lane = threadIdx.x & 31;
    const int wave = (blockIdx.x * blockDim.x + threadIdx.x) >> 5;
    const int i0   = wave * 16;          // first source row of this wave
    const int half = lane >> 4;          // 0 = lanes 0-15, 1 = lanes 16-31
    const int l16  = lane & 15;

    // A tile: two lanes share a row; K3 is the constant 1.0 column.
    float4 ap = Apack[i0 + l16];
    v2f a;
    a.x = half ? ap.z : ap.x;            // K2(tp_z) : K0(tp_x)
    a.y = half ? 1.0f : ap.y;            // K3(1.0)  : K1(tp_y)

    v8f runmin;
#pragma unroll
    for (int r = 0; r < 8; ++r) runmin[r] = 3.402823466e+38f;

#pragma unroll 4
    for (int j0 = 0; j0 < MPTS; j0 += 16) {
        float4 bp = Bpack[j0 + l16];     // one column per lane-pair
        v2f b;
        b.x = half ? bp.z : bp.x;        // K2(-2z)      : K0(-2x)
        b.y = half ? bp.w : bp.y;        // K3(||y||^2)  : K1(-2y)

        v8f c = {};                      // inline-zero SRC2
        v8f d = __builtin_amdgcn_wmma_f32_16x16x4_f32(
            /*neg_a=*/false, a, /*neg_b=*/false, b,
            /*c_mod=*/(short)0, c, /*reuse_a=*/false, /*reuse_b=*/false);

#pragma unroll
        for (int r = 0; r < 8; ++r) runmin[r] = fminf(runmin[r], d[r]);
    }

    // Min across the 16 lanes of each half (columns of the tile).
#pragma unroll
    for (int m = 8; m >= 1; m >>= 1) {
#pragma unroll
        for (int r = 0; r < 8; ++r) {
            float o = __shfl_xor(runmin[r], m, 32);
            runmin[r] = fminf(runmin[r], o);
        }
    }

    // lane 0 holds rows i0..i0+7, lane 16 holds rows i0+8..i0+15.
    // Stored WITHOUT the ||tp_i||^2 term (added in finalize).
    if (l16 == 0) {
#pragma unroll
        for (int r = 0; r < 8; ++r)
            rowmin[i0 + half * 8 + r] = runmin[r];
    }
}

// ---------------------------------------------------------------------------
// Kernel 3: deterministic single-block reduction -> loss scalar.
// Adds back the per-row ||tp_i||^2 (Apack[i].w) hoisted out of the WMMA.
// ---------------------------------------------------------------------------
__global__ void __launch_bounds__(256)
pca_finalize_kernel(const float* __restrict__ rowmin,
                    const float4* __restrict__ Apack,
                    const float* __restrict__ scale,
                    float* __restrict__ out) {
    __shared__ float sm[256];
    float acc = 0.0f;
    for (int i = threadIdx.x; i < NPTS; i += 256)
        acc += rowmin[i] + Apack[i].w;
    sm[threadIdx.x] = acc;
    __syncthreads();
    for (int s = 128; s > 0; s >>= 1) {
        if (threadIdx.x < s) sm[threadIdx.x] += sm[threadIdx.x + s];
        __syncthreads();
    }
    if (threadIdx.x == 0) {
        float sc = scale[0];
        out[0] = sm[0] / (float)NPTS + 0.1f * fmaxf(-sc, 0.0f);
    }
}

// ---------------------------------------------------------------------------
// Host-side launcher (graph-capture safe: only kernel launches on `stream`).
// Workspace layout: [Apack: 256KB][Bpack: 256KB][rowmin: 64KB]
// ---------------------------------------------------------------------------
extern "C" void kernel_launch(void* const* d_in, const int* in_sizes, int n_in,
                              void* d_out, int out_size, void* d_ws, size_t ws_size,
                              hipStream_t stream) {
    const float* src   = (const float*)d_in[0];  // [N,3]
    const float* tgt   = (const float*)d_in[1];  // [M,3]
    const float* scale = (const float*)d_in[2];  // [1]
    const float* trans = (const float*)d_in[3];  // [3]
    float* out = (float*)d_out;

    char* ws = (char*)d_ws;
    float4* Apack = (float4*)(ws);
    float4* Bpack = (float4*)(ws + (size_t)NPTS * sizeof(float4));
    float*  rowmin = (float*)(ws + (size_t)2 * NPTS * sizeof(float4));

    // 1) pack A/B tiles
    pca_prep_kernel<<<(NPTS + 255) / 256, 256, 0, stream>>>(
        src, tgt, scale, trans, Apack, Bpack);

    // 2) WMMA NN sweep: 1024 waves = 128 blocks x 256 threads (8 waves/block)
    pca_nn_kernel<<<(NPTS / 16) / 8, 256, 0, stream>>>(Apack, Bpack, rowmin);

    // 3) reduce to loss
    pca_finalize_kernel<<<1, 256, 0, stream>>>(rowmin, Apack, scale, out);
}